// Anisotropic_SWEncoder_57638461112691
// MI455X (gfx1250) — compile-verified
//
#include <hip/hip_runtime.h>
#include <hip/hip_bf16.h>

typedef __attribute__((ext_vector_type(2))) float v2f;
typedef __attribute__((ext_vector_type(8))) float v8f;

#define DIMF   128              // D
#define NITER  4                // I+1
#define NPROJ  64               // P
#define NQ     100              // Q
#define NCOL   (NITER * NPROJ)  // 256
#define MAXSEG 1024             // >= max segment length (mean 200, sd ~14)
#define SCALE  0.0125f          // 1 / (Q*P)^(1/2) = 1/80
#define BSTRIDE 72              // LDS row stride (words) for proj tile: bank-conflict-free

// ---------------------------------------------------------------------------
// Phase 0: segment starts/counts from sorted batch ids (binary search per g)
// ---------------------------------------------------------------------------
__global__ void seg_bounds_kernel(const int* __restrict__ batch, int n, int g_total,
                                  int* __restrict__ starts, int* __restrict__ counts) {
    int g = blockIdx.x * blockDim.x + threadIdx.x;
    if (g >= g_total) return;
    int lo = 0, hi = n;
    while (lo < hi) { int mid = (lo + hi) >> 1; if (batch[mid] < g) lo = mid + 1; else hi = mid; }
    int s = lo;
    hi = n;
    while (lo < hi) { int mid = (lo + hi) >> 1; if (batch[mid] < g + 1) lo = mid + 1; else hi = mid; }
    starts[g] = s;
    counts[g] = lo - s;
}

// ---------------------------------------------------------------------------
// Phase 1: xp_t[col][n] = x[n, it, :] . proj[:, p]   via V_WMMA_F32_16X16X4_F32
// proj staged in LDS (row stride 72 words -> conflict-free stage + fragment
// reads). One wave per (16-row tile, iteration); all 4 N-tiles accumulated
// in-wave so the HBM-bound x stream is read exactly once. A-fragment loads
// are software-pipelined one k-step ahead to overlap the WMMAs.
// ---------------------------------------------------------------------------
__global__ void __launch_bounds__(128)
proj_wmma_kernel(const float* __restrict__ x,
                 const float* __restrict__ proj,
                 float* __restrict__ xpt, int n) {
    __shared__ float bshm[DIMF * BSTRIDE];

    // cooperative stage: coalesced global read, conflict-free LDS write
    for (int idx = threadIdx.x; idx < DIMF * NPROJ; idx += blockDim.x) {
        const int k = idx >> 6;
        const int c = idx & 63;
        bshm[k * BSTRIDE + c] = proj[idx];
    }
    __syncthreads();

    const int wave  = blockIdx.x * (blockDim.x >> 5) + (threadIdx.x >> 5);
    const int lane  = threadIdx.x & 31;
    const int rtile = wave >> 2;    // 16-row tile index
    const int it    = wave & 3;     // WL iteration 0..3
    const int lrow  = lane & 15;
    const int khalf = lane >> 4;    // 0 or 1
    const int row_tiles = (n + 15) >> 4;

    if (rtile < row_tiles) {        // wave-uniform: EXEC stays all-ones
        int arow = rtile * 16 + lrow;
        if (arow > n - 1) arow = n - 1;                 // clamp, keep all lanes live
        const size_t xbase = (size_t)arow * (NITER * DIMF) + (size_t)it * DIMF;
        const int kofs = khalf * 2;

        v8f c[4] = {};                                  // four 16x16 accumulators

        v2f a;                                          // pipelined A fragment
        a.x = x[xbase + kofs];
        a.y = x[xbase + kofs + 1];

#pragma unroll 4
        for (int k = 0; k < DIMF; k += 4) {
            const int kk = k + kofs;
            v2f an = a;                                 // prefetch next A fragment
            if (k + 4 < DIMF) {
                an.x = x[xbase + kk + 4];
                an.y = x[xbase + kk + 5];
            }
#pragma unroll
            for (int pt = 0; pt < 4; ++pt) {
                const int col = pt * 16 + lrow;
                v2f b;
                b.x = bshm[kk * BSTRIDE + col];
                b.y = bshm[(kk + 1) * BSTRIDE + col];
                c[pt] = __builtin_amdgcn_wmma_f32_16x16x4_f32(
                    /*neg_a=*/false, a, /*neg_b=*/false, b,
                    /*c_mod=*/(short)0, c[pt], /*reuse_a=*/false, /*reuse_b=*/false);
            }
            a = an;
        }

#pragma unroll
        for (int pt = 0; pt < 4; ++pt) {
#pragma unroll
            for (int r = 0; r < 8; ++r) {
                const int row = rtile * 16 + khalf * 8 + r;
                if (row < n) {
                    const int col = it * NPROJ + pt * 16 + lrow;
                    // transposed store: per-lane 8 consecutive rows write-combine in L2;
                    // makes phase-2 segment reads fully coalesced.
                    xpt[(size_t)col * n + row] = c[pt][r];
                }
            }
        }
    }
}

// ---------------------------------------------------------------------------
// Phase 2: per (graph, column): LDS bitonic sort of the segment, then gather
// Q quantiles with "lower" interpolation, scale by 1/80, write [g][i][q][p].
// ---------------------------------------------------------------------------
__global__ void __launch_bounds__(256)
sort_quant_kernel(const float* __restrict__ xpt,
                  const int* __restrict__ starts, const int* __restrict__ counts,
                  const float* __restrict__ cw, float* __restrict__ out, int n) {
    const int col = blockIdx.x;   // 0..255 (fast axis: blocks of same g run together)
    const int g   = blockIdx.y;
    const int tid = threadIdx.x;

    __shared__ float buf[MAXSEG];

    const int cnt   = counts[g];
    const int start = starts[g];
    const int cc    = cnt < MAXSEG ? cnt : MAXSEG;

    int m = 1;                    // next pow2 >= cc (m=1 when cc<=1)
    while (m < cc) m <<= 1;

    const float* src = xpt + (size_t)col * n + start;   // contiguous segment
    for (int j = tid; j < m; j += blockDim.x)
        buf[j] = (j < cc) ? src[j] : 3.402823466e+38f;
    __syncthreads();

    // bitonic sort, ascending
    for (int ksz = 2; ksz <= m; ksz <<= 1) {
        for (int jj = ksz >> 1; jj > 0; jj >>= 1) {
            for (int l = tid; l < m; l += blockDim.x) {
                const int ixj = l ^ jj;
                if (ixj > l) {
                    const float va = buf[l];
                    const float vb = buf[ixj];
                    const bool up = ((l & ksz) == 0);
                    if ((va > vb) == up) { buf[l] = vb; buf[ixj] = va; }
                }
            }
            __syncthreads();
        }
    }

    const int it = col >> 6;
    const int p  = col & 63;
    for (int q = tid; q < NQ; q += blockDim.x) {
        float val = 0.0f;
        if (cc > 0) {
            int idx = (int)(cw[q] * (float)(cc - 1));   // floor, "lower" interp
            if (idx < 0) idx = 0;
            if (idx > cc - 1) idx = cc - 1;
            val = buf[idx] * SCALE;
        }
        out[(((size_t)g * NITER + it) * NQ + q) * NPROJ + p] = val;
    }
}

// ---------------------------------------------------------------------------
extern "C" void kernel_launch(void* const* d_in, const int* in_sizes, int n_in,
                              void* d_out, int out_size, void* d_ws, size_t ws_size,
                              hipStream_t stream) {
    const float* x     = (const float*)d_in[0];
    const int*   batch = (const int*)d_in[1];
    const float* proj  = (const float*)d_in[2];
    const float* cw    = (const float*)d_in[3];
    float*       out   = (float*)d_out;

    const int n = in_sizes[1];                       // N (batch has N entries)
    const int g_total = out_size / (NITER * NQ * NPROJ);

    // workspace layout: xp_t [NCOL * n] floats, then starts[G], counts[G]
    float* xpt = (float*)d_ws;
    size_t xpt_bytes = (size_t)NCOL * (size_t)n * sizeof(float);
    xpt_bytes = (xpt_bytes + 255) & ~(size_t)255;
    int* starts = (int*)((char*)d_ws + xpt_bytes);
    int* counts = starts + g_total;

    // Phase 0: segment bounds
    {
        int blocks = (g_total + 255) / 256;
        seg_bounds_kernel<<<blocks, 256, 0, stream>>>(batch, n, g_total, starts, counts);
    }

    // Phase 1: WMMA projection, transposed store
    {
        const int row_tiles   = (n + 15) / 16;
        const int total_waves = row_tiles * NITER;       // one wave per (tile, iter)
        const int waves_per_block = 4;                   // 128 threads
        const int blocks = (total_waves + waves_per_block - 1) / waves_per_block;
        proj_wmma_kernel<<<blocks, waves_per_block * 32, 0, stream>>>(x, proj, xpt, n);
    }

    // Phase 2: segmented sort + quantile gather
    {
        dim3 grid(NCOL, g_total);
        sort_quant_kernel<<<grid, 256, 0, stream>>>(xpt, starts, counts, cw, out, n);
    }
}